// Lfm2Attention_68745246539786
// MI455X (gfx1250) — compile-verified
//
#include <hip/hip_runtime.h>
#include <hip/hip_bf16.h>
#include <math.h>

// Problem constants (fixed by the reference)
#define T_     2048
#define HID_   2048
#define NH_Q   32
#define NH_K   8
#define HD_    64
#define QKV_N  3072   // (32 + 2*8) * 64
#define NHEADS (NH_Q + NH_K)

typedef __attribute__((ext_vector_type(2))) float v2f;
typedef __attribute__((ext_vector_type(8))) float v8f;

// D = A(16x4) * B(4x16) + C  -- full fp32 WMMA (matches fp32 reference numerics)
__device__ __forceinline__ v8f wmma4(v2f a, v2f b, v8f c) {
  return __builtin_amdgcn_wmma_f32_16x16x4_f32(
      /*neg_a=*/false, a, /*neg_b=*/false, b,
      /*c_mod=*/(short)0, c, /*reuse_a=*/false, /*reuse_b=*/false);
}

// CDNA5 async global->LDS copy (ASYNCcnt path). lds = 32-bit LDS byte offset.
__device__ __forceinline__ void async_copy_b128(unsigned lds, const float* gp) {
  asm volatile("global_load_async_to_lds_b128 %0, %1, off"
               :: "v"(lds), "v"(gp) : "memory");
}
__device__ __forceinline__ void async_wait0() {
  asm volatile("s_wait_asynccnt 0x0" ::: "memory");
}
__device__ __forceinline__ unsigned lds_off(const void* p) {
  return (unsigned)(size_t)p;   // low 32 bits of generic LDS address = DS offset
}

// ---------------------------------------------------------------------------
// fp32 GEMM: C[M,N] = A[M,K] @ B[K,N].
// 256 threads = 8 waves (2x4); WG tile 64(M) x 128(N); wave tile 32x32
// (4 independent WMMA accumulation chains). K staged in 32-deep LDS chunks:
//   A tile 64x32 copied with global_load_async_to_lds_b128,
//   B tile 32x128 transposed into Bt[n][k] so B frags are single b64 loads.
// All fragments of a chunk preloaded, then a 32-WMMA burst.
// ---------------------------------------------------------------------------
__global__ __launch_bounds__(256) void gemm_f32_wmma(
    const float* __restrict__ A, const float* __restrict__ B,
    float* __restrict__ C, int M, int N, int K) {
  __shared__ float As[64 * 36];    // [m][k], stride 36
  __shared__ float Bt[128 * 36];   // [n][k], stride 36 (transposed)

  const int tid  = threadIdx.x;
  const int lane = tid & 31;
  const int wave = tid >> 5;     // 0..7
  const int wm = wave >> 2;      // 0..1 -> 32-row M sub-block
  const int wn = wave & 3;       // 0..3 -> 32-col N sub-block
  const int m0 = blockIdx.y * 64;
  const int n0 = blockIdx.x * 128;

  const int rm = lane & 15;
  const int dd = (lane >> 4) << 1;   // K sub-offset 0 or 2

  const int ar  = tid >> 3;          // A copy: rows ar, ar+32
  const int ac4 = (tid & 7) << 2;    //          16B columns
  const int bn4 = tid & 31;          // B copy: float4 column
  const int bk  = tid >> 5;          //          k rows bk + 8*i

  const v8f zero = {0.f, 0.f, 0.f, 0.f, 0.f, 0.f, 0.f, 0.f};
  v8f acc[2][2] = {{zero, zero}, {zero, zero}};

  for (int kc = 0; kc < K; kc += 32) {
    // ---- A tile: async global -> LDS (no VGPR round trip) ----
    {
      const float* ga = A + (size_t)(m0 + ar) * K + kc + ac4;
      async_copy_b128(lds_off(&As[ar * 36 + ac4]), ga);
      async_copy_b128(lds_off(&As[(ar + 32) * 36 + ac4]), ga + (size_t)32 * K);
    }
    // ---- B tile: load + transpose into Bt[n][k] ----
#pragma unroll
    for (int i = 0; i < 4; ++i) {
      int k = bk + (i << 3);
      float4 v = *(const float4*)(B + (size_t)(kc + k) * N + n0 + (bn4 << 2));
      Bt[(bn4 * 4 + 0) * 36 + k] = v.x;
      Bt[(bn4 * 4 + 1) * 36 + k] = v.y;
      Bt[(bn4 * 4 + 2) * 36 + k] = v.z;
      Bt[(bn4 * 4 + 3) * 36 + k] = v.w;
    }
    async_wait0();
    __syncthreads();

    // ---- preload all fragments for this chunk, then WMMA burst ----
    v2f af[2][8], bf[2][8];
#pragma unroll
    for (int s = 0; s < 8; ++s) {
      af[0][s] = *(const v2f*)&As[(wm * 32 + rm)      * 36 + s * 4 + dd];
      af[1][s] = *(const v2f*)&As[(wm * 32 + 16 + rm) * 36 + s * 4 + dd];
      bf[0][s] = *(const v2f*)&Bt[(wn * 32 + rm)      * 36 + s * 4 + dd];
      bf[1][s] = *(const v2f*)&Bt[(wn * 32 + 16 + rm) * 36 + s * 4 + dd];
    }
#pragma unroll
    for (int s = 0; s < 8; ++s) {
      acc[0][0] = wmma4(af[0][s], bf[0][s], acc[0][0]);
      acc[0][1] = wmma4(af[0][s], bf[1][s], acc[0][1]);
      acc[1][0] = wmma4(af[1][s], bf[0][s], acc[1][0]);
      acc[1][1] = wmma4(af[1][s], bf[1][s], acc[1][1]);
    }
    __syncthreads();
  }

#pragma unroll
  for (int mi = 0; mi < 2; ++mi)
#pragma unroll
    for (int ni = 0; ni < 2; ++ni)
#pragma unroll
      for (int r = 0; r < 8; ++r) {
        int m = m0 + wm * 32 + mi * 16 + r + ((lane >> 4) << 3);
        C[(size_t)m * N + n0 + wn * 32 + ni * 16 + rm] = acc[mi][ni][r];
      }
}

// ---------------------------------------------------------------------------
// Per-(t, head) RMSNorm + NeoX RoPE, in place on qkv. One wave per head-row.
// ---------------------------------------------------------------------------
__global__ __launch_bounds__(256) void norm_rope(
    const int* __restrict__ positions, float* __restrict__ qkv,
    const float* __restrict__ qw, const float* __restrict__ kw) {
  const int lane = threadIdx.x & 31;
  const int gw   = (blockIdx.x * blockDim.x + threadIdx.x) >> 5;
  const int t = gw / NHEADS;
  const int h = gw - t * NHEADS;
  const float* w = (h < NH_Q) ? qw : kw;
  float* x = qkv + (size_t)t * QKV_N + h * HD_;

  float e0 = x[lane], e1 = x[lane + 32];
  float ss = e0 * e0 + e1 * e1;
#pragma unroll
  for (int m = 16; m >= 1; m >>= 1) ss += __shfl_xor(ss, m, 32);
  float r = rsqrtf(ss * (1.f / 64.f) + 1e-5f);
  float y0 = e0 * r * w[lane];
  float y1 = e1 * r * w[lane + 32];

  float pos  = (float)positions[t];
  float invf = powf(10000.f, -(float)lane * (1.f / 32.f));
  float fr = pos * invf;
  float c = cosf(fr), s = sinf(fr);
  x[lane]      = y0 * c - y1 * s;
  x[lane + 32] = y1 * c + y0 * s;
}

// ---------------------------------------------------------------------------
// Flash attention. 128 threads = 4 waves; the 4 waves of a WG share the same
// query block AND the same kv head (h>>2 uniform), so K/V tiles are staged
// cooperatively in LDS once per key block:
//   K tile 16x64 via global_load_async_to_lds_b128 (frag = b64 load),
//   V tile transposed Vt[d][k] (frag = b64 load).
// One wave per (16-query block, q head); online softmax; P re-laid out to
// A-fragments via a per-wave LDS bounce.
// ---------------------------------------------------------------------------
__global__ __launch_bounds__(128) void attn_fwd(
    const float* __restrict__ qkv, float* __restrict__ attn) {
  __shared__ float Kt[16 * 68];        // [key][d]
  __shared__ float Vt[64 * 18];        // [d][key] (transposed)
  __shared__ float plds[4][16 * 17];   // per-wave P bounce

  const int tid   = threadIdx.x;
  const int lane  = tid & 31;
  const int wslot = tid >> 5;                // 0..3
  const int gw = blockIdx.x * 4 + wslot;     // 0..4095
  const int h  = gw & (NH_Q - 1);
  const int qb = gw >> 5;                    // uniform across WG
  const int kh = h >> 2;                     // uniform across WG (GQA 4:1)
  const int rm = lane & 15;
  const int hi = lane >> 4;
  const int dd = hi << 1;

  const v8f zero = {0.f, 0.f, 0.f, 0.f, 0.f, 0.f, 0.f, 0.f};

  // Q tile (16x64) resident in registers in A-fragment layout, scale folded in.
  v2f qa[16];
  {
    const float* qbase = qkv + (size_t)(qb * 16 + rm) * QKV_N + h * HD_;
#pragma unroll
    for (int s = 0; s < 16; ++s) {
      v2f q = *(const v2f*)(qbase + s * 4 + dd);
      qa[s].x = q.x * 0.125f;  // D^-0.5
      qa[s].y = q.y * 0.125f;
    }
  }

  v8f Ot[4] = {zero, zero, zero, zero};
  float rowmax[8], rowsum[8];
#pragma unroll
  for (int r = 0; r < 8; ++r) { rowmax[r] = -__builtin_inff(); rowsum[r] = 0.f; }

  float* pl = plds[wslot];

  for (int kb = 0; kb <= qb; ++kb) {
    // ---- cooperative K/V tile staging (128 threads) ----
#pragma unroll
    for (int i = 0; i < 2; ++i) {
      int idx = tid * 2 + i;                 // 0..255 -> 256 float4s per tile
      int r = idx >> 4, c4 = (idx & 15) << 2;
      const float* row = qkv + (size_t)(kb * 16 + r) * QKV_N;
      async_copy_b128(lds_off(&Kt[r * 68 + c4]), row + HID_ + kh * HD_ + c4);
      float4 v = *(const float4*)(row + HID_ + NH_K * HD_ + kh * HD_ + c4);
      Vt[(c4 + 0) * 18 + r] = v.x;
      Vt[(c4 + 1) * 18 + r] = v.y;
      Vt[(c4 + 2) * 18 + r] = v.z;
      Vt[(c4 + 3) * 18 + r] = v.w;
    }
    async_wait0();
    __syncthreads();

    // ---- S = Q K^T : preload 16 B-frags, two interleaved WMMA chains ----
    v2f kf[16];
#pragma unroll
    for (int s = 0; s < 16; ++s) kf[s] = *(const v2f*)&Kt[rm * 68 + s * 4 + dd];
    v8f S0 = zero, S1 = zero;
#pragma unroll
    for (int s = 0; s < 8; ++s) {
      S0 = wmma4(qa[2 * s],     kf[2 * s],     S0);
      S1 = wmma4(qa[2 * s + 1], kf[2 * s + 1], S1);
    }
    v8f S;
#pragma unroll
    for (int r = 0; r < 8; ++r) S[r] = S0[r] + S1[r];

    // ---- causal mask on the diagonal block (selects only) ----
    if (kb == qb) {
#pragma unroll
      for (int r = 0; r < 8; ++r) {
        int mrow = r + (hi << 3);
        S[r] = (rm > mrow) ? -3.0e38f : S[r];
      }
    }

    // ---- online softmax (row = 16 lanes of one half-wave) ----
    float al[8];
#pragma unroll
    for (int r = 0; r < 8; ++r) {
      float mx = S[r];
#pragma unroll
      for (int m = 8; m >= 1; m >>= 1) mx = fmaxf(mx, __shfl_xor(mx, m, 32));
      float nm = fmaxf(rowmax[r], mx);
      float a  = __expf(rowmax[r] - nm);
      float p  = __expf(S[r] - nm);
      float ps = p;
#pragma unroll
      for (int m = 8; m >= 1; m >>= 1) ps += __shfl_xor(ps, m, 32);
      rowsum[r] = rowsum[r] * a + ps;
      rowmax[r] = nm;
      al[r] = a;
      S[r]  = p;
    }
#pragma unroll
    for (int i = 0; i < 4; ++i)
#pragma unroll
      for (int r = 0; r < 8; ++r) Ot[i][r] *= al[r];

    // ---- P: C-layout -> LDS -> A-layout (wave-internal) ----
#pragma unroll
    for (int r = 0; r < 8; ++r) pl[(r + (hi << 3)) * 17 + rm] = S[r];
    v2f pa[4];
#pragma unroll
    for (int s = 0; s < 4; ++s) {
      pa[s].x = pl[rm * 17 + s * 4 + dd];
      pa[s].y = pl[rm * 17 + s * 4 + dd + 1];
    }

    // ---- O += P V : preload 16 B-frags, 4 independent chains ----
    v2f vf[4][4];
#pragma unroll
    for (int nt = 0; nt < 4; ++nt)
#pragma unroll
      for (int s = 0; s < 4; ++s)
        vf[nt][s] = *(const v2f*)&Vt[(nt * 16 + rm) * 18 + s * 4 + dd];
#pragma unroll
    for (int s = 0; s < 4; ++s) {
      Ot[0] = wmma4(pa[s], vf[0][s], Ot[0]);
      Ot[1] = wmma4(pa[s], vf[1][s], Ot[1]);
      Ot[2] = wmma4(pa[s], vf[2][s], Ot[2]);
      Ot[3] = wmma4(pa[s], vf[3][s], Ot[3]);
    }
    __syncthreads();   // protect K/V tiles before next iteration's staging
  }

  // ---- epilogue: normalize and store ----
#pragma unroll
  for (int r = 0; r < 8; ++r) {
    float inv = 1.f / rowsum[r];
    int trow = qb * 16 + r + (hi << 3);
    float* op = attn + (size_t)trow * (NH_Q * HD_) + h * HD_;
#pragma unroll
    for (int nt = 0; nt < 4; ++nt) op[nt * 16 + rm] = Ot[nt][r] * inv;
  }
}

// ---------------------------------------------------------------------------
extern "C" void kernel_launch(void* const* d_in, const int* in_sizes, int n_in,
                              void* d_out, int out_size, void* d_ws, size_t ws_size,
                              hipStream_t stream) {
  const int*   positions = (const int*)  d_in[0];
  const float* hidden    = (const float*)d_in[1];
  const float* Wqkv      = (const float*)d_in[2];
  const float* Wout      = (const float*)d_in[3];
  const float* qw        = (const float*)d_in[4];
  const float* kw        = (const float*)d_in[5];
  float* out = (float*)d_out;

  float* qkv  = (float*)d_ws;                  // T x 3072 fp32 (25.2 MB)
  float* attn = qkv + (size_t)T_ * QKV_N;      // T x 2048 fp32 (16.8 MB)

  // 1) qkv = hidden @ Wqkv
  dim3 g1(QKV_N / 128, T_ / 64);
  gemm_f32_wmma<<<g1, 256, 0, stream>>>(hidden, Wqkv, qkv, T_, QKV_N, HID_);

  // 2) RMSNorm + RoPE on q and k head-rows (in place); v passes through
  int waves = T_ * NHEADS;                       // 81920, exact
  norm_rope<<<(waves * 32) / 256, 256, 0, stream>>>(positions, qkv, qw, kw);

  // 3) causal GQA flash attention
  attn_fwd<<<((T_ / 16) * NH_Q) / 4, 128, 0, stream>>>(qkv, attn);

  // 4) out = attn @ Wout
  dim3 g2(HID_ / 128, T_ / 64);
  gemm_f32_wmma<<<g2, 256, 0, stream>>>(attn, Wout, out, T_, HID_, NH_Q * HD_);
}